// NTM_63685775065368
// MI455X (gfx1250) — compile-verified
//
#include <hip/hip_runtime.h>
#include <hip/hip_bf16.h>
#include <math.h>

typedef __attribute__((ext_vector_type(2))) float v2f;
typedef __attribute__((ext_vector_type(8))) float v8f;

#define NROWS 65536
#define MDIM  512
#define HDIM  1024
#define DINSZ 512
#define EPSV  1e-8f

// ---- workspace layout (float offsets) ----
#define WS_GATES 0                     // 4096
#define WS_KVEC  4096                  // 512
#define WS_KN    (4096 + 512)          // 512
#define WS_ERASE (4096 + 1024)         // 512
#define WS_ADD   (4096 + 1536)         // 512
#define WS_BETA  (4096 + 2048)         // 1
#define WS_GMAX  (WS_BETA + 1)
#define WS_GSUM  (WS_BETA + 2)
#define WS_T     8192                  // 65536 (t, then e in place)
#define WS_PMAX  (8192 + 65536)        // 1024
#define WS_PSUM  (WS_PMAX + 1024)      // 256
#define WS_PREAD (WS_PSUM + 256)       // 2048 * 512

// ---- d_out layout (float offsets): (out, h, c, mem_new, read_new) ----
#define DO_OUT  0
#define DO_H    512
#define DO_C    (512 + 1024)
#define DO_MEM  (512 + 2048)
#define DO_READ (DO_MEM + (size_t)NROWS * MDIM)

__device__ __forceinline__ float sigf(float x) { return 1.0f / (1.0f + expf(-x)); }
__device__ __forceinline__ float softplusf(float x) {
  return (x > 20.0f) ? x : log1pf(expf(x));
}
__device__ __forceinline__ float dot4(const float4 a, const float4 b, float acc) {
  acc = fmaf(a.x, b.x, acc);
  acc = fmaf(a.y, b.y, acc);
  acc = fmaf(a.z, b.z, acc);
  acc = fmaf(a.w, b.w, acc);
  return acc;
}

// ---------------- K1: gates = W_ih@[x;read] + b_ih + W_hh@h0 + b_hh ----------------
__global__ void __launch_bounds__(256) k_gates(const float* __restrict__ x,
                                               const float* __restrict__ rv,
                                               const float* __restrict__ h0,
                                               const float* __restrict__ Wih,
                                               const float* __restrict__ bih,
                                               const float* __restrict__ Whh,
                                               const float* __restrict__ bhh,
                                               float* __restrict__ gates) {
  const int lane = threadIdx.x & 31;
  const int wave = threadIdx.x >> 5;
  const int r = blockIdx.x * 8 + wave;            // 512 blocks * 8 waves = 4096 rows
  const float4* wi4 = (const float4*)(Wih + (size_t)r * (DINSZ + MDIM));
  const float4* wh4 = (const float4*)(Whh + (size_t)r * HDIM);
  const float4* x4  = (const float4*)x;           // 128 float4
  const float4* rv4 = (const float4*)rv;          // 128 float4
  const float4* h04 = (const float4*)h0;          // 256 float4
  float acc = 0.f;
#pragma unroll
  for (int it = 0; it < 4; ++it) {                // xi[0:512) = x
    const int q = it * 32 + lane;
    acc = dot4(wi4[q], x4[q], acc);
  }
#pragma unroll
  for (int it = 0; it < 4; ++it) {                // xi[512:1024) = read_vec
    const int q = 128 + it * 32 + lane;
    acc = dot4(wi4[q], rv4[q - 128], acc);
  }
#pragma unroll
  for (int it = 0; it < 8; ++it) {                // W_hh @ h0
    const int q = it * 32 + lane;
    acc = dot4(wh4[q], h04[q], acc);
  }
  for (int off = 16; off; off >>= 1) acc += __shfl_xor(acc, off, 32);
  if (lane == 0) gates[r] = acc + bih[r] + bhh[r];
}

// ---------------- K2: LSTM cell elementwise -> h, c ----------------
__global__ void k_cell(const float* __restrict__ gates, const float* __restrict__ c0,
                       float* __restrict__ outp) {
  const int j = threadIdx.x;  // 1024 threads, 1 block
  float ig = sigf(gates[j]);
  float fg = sigf(gates[HDIM + j]);
  float gg = tanhf(gates[2 * HDIM + j]);
  float og = sigf(gates[3 * HDIM + j]);
  float c = fg * c0[j] + ig * gg;
  float h = og * tanhf(c);
  outp[DO_H + j] = h;
  outp[DO_C + j] = c;
}

// ---------------- K3: head matvecs over h ----------------
__global__ void __launch_bounds__(256) k_heads(const float* __restrict__ hvec,
                                               const float* __restrict__ Wfc, const float* __restrict__ bfc,
                                               const float* __restrict__ Wkey, const float* __restrict__ bkey,
                                               const float* __restrict__ Wbeta, const float* __restrict__ bbeta,
                                               const float* __restrict__ Werase, const float* __restrict__ berase,
                                               const float* __restrict__ Wadd, const float* __restrict__ badd,
                                               float* __restrict__ d_out, float* __restrict__ ws) {
  const int lane = threadIdx.x & 31;
  const int wave = threadIdx.x >> 5;
  const int row = blockIdx.x * 8 + wave;          // 257 blocks -> rows 0..2055
  if (row > 2048) return;
  const float* W;
  float bias;
  float* dst;
  int act;
  if (row < 512)        { W = Wfc    + (size_t)row * HDIM;          bias = bfc[row];           dst = d_out + DO_OUT + row;          act = 0; }
  else if (row < 1024)  { W = Wkey   + (size_t)(row - 512) * HDIM;  bias = bkey[row - 512];    dst = ws + WS_KVEC + (row - 512);    act = 1; }
  else if (row < 1536)  { W = Werase + (size_t)(row - 1024) * HDIM; bias = berase[row - 1024]; dst = ws + WS_ERASE + (row - 1024);  act = 0; }
  else if (row < 2048)  { W = Wadd   + (size_t)(row - 1536) * HDIM; bias = badd[row - 1536];   dst = ws + WS_ADD + (row - 1536);    act = 1; }
  else                  { W = Wbeta;                                bias = bbeta[0];           dst = ws + WS_BETA;                  act = 2; }
  const float4* W4 = (const float4*)W;
  const float4* h4 = (const float4*)hvec;
  float acc = 0.f;
#pragma unroll
  for (int it = 0; it < 8; ++it) {
    const int q = it * 32 + lane;
    acc = dot4(W4[q], h4[q], acc);
  }
  for (int off = 16; off; off >>= 1) acc += __shfl_xor(acc, off, 32);
  if (lane == 0) {
    float v = acc + bias;
    if (act == 0) v = sigf(v);
    else if (act == 1) v = tanhf(v);
    else v = softplusf(v) + EPSV;
    *dst = v;
  }
}

// ---------------- K4: kn = kvec / (||kvec|| + eps) ----------------
__global__ void k_keynorm(float* __restrict__ ws) {
  __shared__ float red[512];
  const int t = threadIdx.x;  // 512
  float v = ws[WS_KVEC + t];
  red[t] = v * v;
  __syncthreads();
  for (int s = 256; s; s >>= 1) {
    if (t < s) red[t] += red[t + s];
    __syncthreads();
  }
  ws[WS_KN + t] = v / (sqrtf(red[0]) + EPSV);
}

// ---------------- K5: similarity via V_WMMA_F32_16X16X4_F32 ----------------
// Each block: 64 rows of memory, 4 waves, 16 rows per wave.
// Tiles staged global->LDS with CDNA5 async DMA (global_load_async_to_lds_b128,
// ASYNCcnt drained once per tile), LDS stride 36 floats (144 B): 16 B-aligned
// B128 destinations, bank-conflict-free WMMA A-fragment ds reads.
// A (16x4 f32): lanes 0-15 hold K=0(v0),K=1(v1); lanes 16-31 hold K=2,K=3.
// B (4x16): column 0 = kn chunk -> lane0 {kn[k],kn[k+1]}, lane16 {kn[k+2],kn[k+3]}.
// D column 0: lane0 vgprs 0..7 = rows 0..7, lane16 vgprs 0..7 = rows 8..15.
__global__ void __launch_bounds__(128) k_sim(const float* __restrict__ mem,
                                             float* __restrict__ ws) {
  __shared__ float tileA[64][36];
  __shared__ float knS[512];
  __shared__ float wDots[4][16];
  __shared__ float waveMax[4];
  const int tid = threadIdx.x;
  const int lane = tid & 31;
  const int wave = tid >> 5;
  const int nBase = blockIdx.x * 64;

  for (int j = tid; j < 512; j += 128) knS[j] = ws[WS_KN + j];
  const float betaV = ws[WS_BETA];
  v8f acc = {0.f, 0.f, 0.f, 0.f, 0.f, 0.f, 0.f, 0.f};
  float sq = 0.f;
  __syncthreads();

  const int r = wave * 16 + (lane & 15);
  const int koff = (lane >= 16) ? 2 : 0;
  const int lrow = tid >> 3;            // 0..15
  const int lcol = (tid & 7) * 4;       // 0,4,...,28 (floats)

  for (int kc = 0; kc < 512; kc += 32) {
    // async global -> LDS: 4 x (b128 per lane); uniform trip count, no exec masking
#pragma unroll
    for (int it = 0; it < 4; ++it) {
      const int row = lrow + it * 16;
      const unsigned long long gaddr =
          (unsigned long long)(uintptr_t)(mem + (size_t)(nBase + row) * MDIM + kc + lcol);
      const unsigned ldsOff = (unsigned)(uintptr_t)(&tileA[row][lcol]);
      asm volatile("global_load_async_to_lds_b128 %0, %1, off"
                   :: "v"(ldsOff), "v"(gaddr) : "memory");
    }
    asm volatile("s_wait_asynccnt 0x0" ::: "memory");
    __syncthreads();
#pragma unroll
    for (int s = 0; s < 8; s++) {
      const int k = s * 4 + koff;
      const float a0 = tileA[r][k];
      const float a1 = tileA[r][k + 1];
      sq = fmaf(a0, a0, fmaf(a1, a1, sq));
      const int kg = kc + s * 4;
      const float b0 = (lane == 0) ? knS[kg]     : ((lane == 16) ? knS[kg + 2] : 0.f);
      const float b1 = (lane == 0) ? knS[kg + 1] : ((lane == 16) ? knS[kg + 3] : 0.f);
      v2f A = {a0, a1};
      v2f B = {b0, b1};
      acc = __builtin_amdgcn_wmma_f32_16x16x4_f32(false, A, false, B, (short)0, acc,
                                                  false, false);
    }
    __syncthreads();
  }

  if (lane == 0) {
#pragma unroll
    for (int i = 0; i < 8; i++) wDots[wave][i] = acc[i];
  }
  if (lane == 16) {
#pragma unroll
    for (int i = 0; i < 8; i++) wDots[wave][8 + i] = acc[i];
  }
  __syncthreads();

  const float ss = sq + __shfl_xor(sq, 16, 32);   // full row sumsq on lanes 0-15
  float tv = -3.0e38f;
  if (lane < 16) {
    float dot = wDots[wave][lane];
    float t = betaV * (dot / (sqrtf(ss) + EPSV));
    ws[WS_T + nBase + wave * 16 + lane] = t;
    tv = t;
  }
  for (int off = 16; off; off >>= 1) tv = fmaxf(tv, __shfl_xor(tv, off, 32));
  if (lane == 0) waveMax[wave] = tv;
  __syncthreads();
  if (tid == 0) {
    ws[WS_PMAX + blockIdx.x] =
        fmaxf(fmaxf(waveMax[0], waveMax[1]), fmaxf(waveMax[2], waveMax[3]));
  }
}

// ---------------- K6: reduce block maxima ----------------
__global__ void k_rmax(float* __restrict__ ws) {
  __shared__ float red[256];
  const int t = threadIdx.x;
  float m = -3.0e38f;
  for (int i = t; i < 1024; i += 256) m = fmaxf(m, ws[WS_PMAX + i]);
  red[t] = m;
  __syncthreads();
  for (int s = 128; s; s >>= 1) {
    if (t < s) red[t] = fmaxf(red[t], red[t + s]);
    __syncthreads();
  }
  if (t == 0) ws[WS_GMAX] = red[0];
}

// ---------------- K7: e = exp(t - gmax), partial sums ----------------
__global__ void __launch_bounds__(256) k_exp(float* __restrict__ ws) {
  __shared__ float red[256];
  const int t = threadIdx.x;
  const int i = blockIdx.x * 256 + t;  // 256 blocks * 256 = 65536
  const float gmax = ws[WS_GMAX];
  float e = expf(ws[WS_T + i] - gmax);
  ws[WS_T + i] = e;
  red[t] = e;
  __syncthreads();
  for (int s = 128; s; s >>= 1) {
    if (t < s) red[t] += red[t + s];
    __syncthreads();
  }
  if (t == 0) ws[WS_PSUM + blockIdx.x] = red[0];
}

// ---------------- K8: reduce partial sums ----------------
__global__ void k_rsum(float* __restrict__ ws) {
  __shared__ float red[256];
  const int t = threadIdx.x;
  red[t] = ws[WS_PSUM + t];
  __syncthreads();
  for (int s = 128; s; s >>= 1) {
    if (t < s) red[t] += red[t + s];
    __syncthreads();
  }
  if (t == 0) ws[WS_GSUM] = red[0];
}

// ---------------- K9: memory update + per-block partial read vectors ----------------
__global__ void __launch_bounds__(256) k_update(const float* __restrict__ mem,
                                                float* __restrict__ ws,
                                                float* __restrict__ d_out) {
  const int tid = threadIdx.x;
  const int m0 = tid * 2;                          // 256 threads cover 512 cols
  const float inv = 1.0f / ws[WS_GSUM];
  const float e0 = ws[WS_ERASE + m0], e1 = ws[WS_ERASE + m0 + 1];
  const float a0 = ws[WS_ADD + m0],   a1 = ws[WS_ADD + m0 + 1];
  float acc0 = 0.f, acc1 = 0.f;
  const int rbase = blockIdx.x * 32;               // 2048 blocks * 32 rows
  for (int n = 0; n < 32; n++) {
    const int rr = rbase + n;
    const float w = ws[WS_T + rr] * inv;
    const float2 mv = *(const float2*)(mem + (size_t)rr * MDIM + m0);
    float nv0 = mv.x * (1.0f - w * e0) + w * a0;
    float nv1 = mv.y * (1.0f - w * e1) + w * a1;
    float2 o; o.x = nv0; o.y = nv1;
    *(float2*)(d_out + DO_MEM + (size_t)rr * MDIM + m0) = o;
    acc0 = fmaf(w, nv0, acc0);
    acc1 = fmaf(w, nv1, acc1);
  }
  ws[WS_PREAD + (size_t)blockIdx.x * MDIM + m0] = acc0;
  ws[WS_PREAD + (size_t)blockIdx.x * MDIM + m0 + 1] = acc1;
}

// ---------------- K10: deterministic reduce of partial read vectors ----------------
__global__ void k_read(const float* __restrict__ ws, float* __restrict__ d_out) {
  const int m = blockIdx.x * 256 + threadIdx.x;    // 2 blocks * 256 = 512
  float acc = 0.f;
  for (int b = 0; b < 2048; b++) acc += ws[WS_PREAD + (size_t)b * MDIM + m];
  d_out[DO_READ + m] = acc;
}

extern "C" void kernel_launch(void* const* d_in, const int* in_sizes, int n_in,
                              void* d_out, int out_size, void* d_ws, size_t ws_size,
                              hipStream_t stream) {
  (void)in_sizes; (void)n_in; (void)out_size; (void)ws_size;
  const float* x        = (const float*)d_in[0];
  const float* h0       = (const float*)d_in[1];
  const float* c0       = (const float*)d_in[2];
  const float* memory   = (const float*)d_in[3];
  const float* read_vec = (const float*)d_in[4];
  const float* W_ih  = (const float*)d_in[5];
  const float* b_ih  = (const float*)d_in[6];
  const float* W_hh  = (const float*)d_in[7];
  const float* b_hh  = (const float*)d_in[8];
  const float* W_fc  = (const float*)d_in[9];
  const float* b_fc  = (const float*)d_in[10];
  const float* W_key = (const float*)d_in[11];
  const float* b_key = (const float*)d_in[12];
  const float* W_beta  = (const float*)d_in[13];
  const float* b_beta  = (const float*)d_in[14];
  const float* W_erase = (const float*)d_in[15];
  const float* b_erase = (const float*)d_in[16];
  const float* W_add   = (const float*)d_in[17];
  const float* b_add   = (const float*)d_in[18];
  float* outp = (float*)d_out;
  float* ws   = (float*)d_ws;

  k_gates<<<512, 256, 0, stream>>>(x, read_vec, h0, W_ih, b_ih, W_hh, b_hh, ws + WS_GATES);
  k_cell<<<1, 1024, 0, stream>>>(ws + WS_GATES, c0, outp);
  k_heads<<<257, 256, 0, stream>>>(outp + DO_H, W_fc, b_fc, W_key, b_key, W_beta, b_beta,
                                   W_erase, b_erase, W_add, b_add, outp, ws);
  k_keynorm<<<1, 512, 0, stream>>>(ws);
  k_sim<<<1024, 128, 0, stream>>>(memory, ws);
  k_rmax<<<1, 256, 0, stream>>>(ws);
  k_exp<<<256, 256, 0, stream>>>(ws);
  k_rsum<<<1, 256, 0, stream>>>(ws);
  k_update<<<2048, 256, 0, stream>>>(memory, ws, outp);
  k_read<<<2, 256, 0, stream>>>(ws, outp);
}